// AI4DEM_81441169867152
// MI455X (gfx1250) — compile-verified
//
#include <hip/hip_runtime.h>

// ---------------- problem constants (from reference) ----------------
#define D_GRID 200
#define TI 8                  // interior tile edge
#define H  (TI + 4)           // haloed tile edge = 12
#define HH (H * H)            // 144
#define HVOL (H * H * H)      // 1728
#define NF 6                  // x,y,z,vx,vy,vz staged in LDS
#define NT (TI * TI * TI)     // 512 threads / block

__constant__ const float kPS   = 0.1f;        // particle_size
__constant__ const float kKN   = 600000.0f;   // normal stiffness
__constant__ const float kDT   = 1e-4f;
// PM = 4/3 * 3.1415 * 0.05^3 * 2700 = 1.4136750
__constant__ const float kPM   = 1.4136750f;
// ETA = 2*gamma*sqrt(KN*PM), gamma = a/sqrt(a^2+1), a = -ln(0.5)/pi
__constant__ const float kETA  = 396.85748f;
__constant__ const float kDTPM = 7.0737585e-5f; // DT / PM
__constant__ const float kWALL = 9.9f;          // D*CELL - CELL - 0.5*PS (== hi threshold)

__global__ __launch_bounds__(NT) void dem_step_kernel(
    const float* __restrict__ xg,  const float* __restrict__ yg,
    const float* __restrict__ zg,  const float* __restrict__ vxg,
    const float* __restrict__ vyg, const float* __restrict__ vzg,
    const float* __restrict__ mg,  float* __restrict__ out)
{
    __shared__ float smem[NF * HVOL];   // 41472 bytes, SoA per field

    const int tx = threadIdx.x, ty = threadIdx.y, tz = threadIdx.z;
    const int tid = tx + ty * TI + tz * TI * TI;
    const int bi = blockIdx.z, bj = blockIdx.y, bk = blockIdx.x;

    // LDS byte offset of smem: generic LDS pointer low 32 bits == LDS offset
    const unsigned lds_base = (unsigned)(unsigned long long)(const void*)smem;

    const float* srcs[NF] = {xg, yg, zg, vxg, vyg, vzg};

    // ---- stage 12^3 halo tile (periodic wrap) via async global->LDS DMA ----
    for (int f = 0; f < NF; ++f) {
        const float* base = srcs[f];              // uniform -> SGPR pair
        for (int e = tid; e < HVOL; e += NT) {
            int hi = e / HH;
            int r  = e - hi * HH;
            int hj = r / H;
            int hk = r - hj * H;
            int gi = bi * TI + hi - 2; gi += (gi < 0) ? D_GRID : 0; gi -= (gi >= D_GRID) ? D_GRID : 0;
            int gj = bj * TI + hj - 2; gj += (gj < 0) ? D_GRID : 0; gj -= (gj >= D_GRID) ? D_GRID : 0;
            int gk = bk * TI + hk - 2; gk += (gk < 0) ? D_GRID : 0; gk -= (gk >= D_GRID) ? D_GRID : 0;
            unsigned goff = (unsigned)(((gi * D_GRID) + gj) * D_GRID + gk) * 4u;
            unsigned loff = lds_base + (unsigned)(f * HVOL + e) * 4u;
            // GVS mode: copies 4 bytes at base plus offset goff into LDS at loff; tracked by ASYNCcnt
            asm volatile("global_load_async_to_lds_b32 %0, %1, %2"
                         :
                         : "v"(loff), "v"(goff), "s"(base)
                         : "memory");
        }
    }

    // cell index + mask load issued before the wait so VMEM latency hides
    const int gi0 = bi * TI + tz, gj0 = bj * TI + ty, gk0 = bk * TI + tx;
    const int cell = (gi0 * D_GRID + gj0) * D_GRID + gk0;
    const float m = mg[cell];

    asm volatile("s_wait_asynccnt 0" ::: "memory");
    __syncthreads();

    // ---- per-cell 125-tap pair forces from LDS ----
    const int li = tz + 2, lj = ty + 2, lk = tx + 2;
    const int c0 = li * HH + lj * H + lk;

    const float xc  = smem[0 * HVOL + c0];
    const float yc  = smem[1 * HVOL + c0];
    const float zc  = smem[2 * HVOL + c0];
    const float vxc = smem[3 * HVOL + c0];
    const float vyc = smem[4 * HVOL + c0];
    const float vzc = smem[5 * HVOL + c0];

    float fx = 0.0f, fy = 0.0f, fz = 0.0f;

    for (int di = -2; di <= 2; ++di) {
#pragma unroll
        for (int dj = -2; dj <= 2; ++dj) {
            const int rowbase = (li + di) * HH + (lj + dj) * H + lk;
#pragma unroll
            for (int dk = -2; dk <= 2; ++dk) {
                const int n = rowbase + dk;
                const float dx = xc - smem[0 * HVOL + n];
                const float dy = yc - smem[1 * HVOL + n];
                const float dz = zc - smem[2 * HVOL + n];
                const float sq = dx * dx + dy * dy + dz * dz;
                const float dist = __builtin_amdgcn_sqrtf(sq);       // sqrt(0)=0 == _safe_dist
                const float dcl  = fmaxf(dist, 1e-4f);
                const float inv  = __builtin_amdgcn_rcpf(dcl);
                const float dvx = vxc - smem[3 * HVOL + n];
                const float dvy = vyc - smem[4 * HVOL + n];
                const float dvz = vzc - smem[5 * HVOL + n];
                const float vn  = (dvx * dx + dvy * dy + dvz * dz) * inv;
                // original adds each spring+damper twice -> factor 2
                const float coef = (dist < kPS)
                    ? 2.0f * (kKN * (dist - kPS) + kETA * vn) * inv
                    : 0.0f;
                fx += coef * dx;
                fy += coef * dy;
                fz += coef * dz;
            }
        }
    }

    // ---- wall boundary forces + update ----
    const float lo1 = kPS, lo2 = 1.5f * kPS;

    const float wl = (xc > lo1 && xc < lo2) ? 1.0f : 0.0f;
    const float wr = (xc > kWALL) ? 1.0f : 0.0f;
    const float fxb = kKN * wl * m * (lo2 - xc) - kKN * wr * m * (xc - kWALL)
                      - kETA * vxc * (wl + wr) * m;

    const float wb = (yc > lo1 && yc < lo2) ? 1.0f : 0.0f;
    const float wt = (yc > kWALL) ? 1.0f : 0.0f;
    const float fyb = kKN * wb * m * (lo2 - yc) - kKN * wt * m * (yc - kWALL)
                      - kETA * vyc * (wb + wt) * m;

    const float wf = (zc > lo1 && zc < lo2) ? 1.0f : 0.0f;
    const float wk = (zc > kWALL) ? 1.0f : 0.0f;
    const float fzb = kKN * wf * m * (lo2 - zc) - kKN * wk * m * (zc - kWALL)
                      - kETA * vzc * (wf + wk) * m;

    const float vx2 = vxc + kDTPM * m * (-fx + fxb);
    const float vy2 = vyc + kDTPM * m * (-fy + fyb);
    const float vz2 = vzc + kDTPM * m * (-9.8f * kPM - fz + fzb);
    const float x2 = xc + kDT * vx2;
    const float y2 = yc + kDT * vy2;
    const float z2 = zc + kDT * vz2;

    // Non-temporal stores: outputs are write-once; keep the ~224MB input
    // working set resident in the 192MB L2 instead of evicting it.
    const int VOL = D_GRID * D_GRID * D_GRID;
    __builtin_nontemporal_store(x2,  out + 0 * VOL + cell);
    __builtin_nontemporal_store(y2,  out + 1 * VOL + cell);
    __builtin_nontemporal_store(z2,  out + 2 * VOL + cell);
    __builtin_nontemporal_store(vx2, out + 3 * VOL + cell);
    __builtin_nontemporal_store(vy2, out + 4 * VOL + cell);
    __builtin_nontemporal_store(vz2, out + 5 * VOL + cell);
}

extern "C" void kernel_launch(void* const* d_in, const int* in_sizes, int n_in,
                              void* d_out, int out_size, void* d_ws, size_t ws_size,
                              hipStream_t stream) {
    (void)in_sizes; (void)n_in; (void)d_ws; (void)ws_size; (void)out_size;
    const float* x  = (const float*)d_in[0];
    const float* y  = (const float*)d_in[1];
    const float* z  = (const float*)d_in[2];
    const float* vx = (const float*)d_in[3];
    const float* vy = (const float*)d_in[4];
    const float* vz = (const float*)d_in[5];
    const float* mk = (const float*)d_in[6];

    dim3 grid(D_GRID / TI, D_GRID / TI, D_GRID / TI);  // 25^3 blocks, exact fit
    dim3 block(TI, TI, TI);                            // 512 threads = 16 waves
    dem_step_kernel<<<grid, block, 0, stream>>>(x, y, z, vx, vy, vz, mk,
                                                (float*)d_out);
}